// MaskLoss_40913858462283
// MI455X (gfx1250) — compile-verified
//
#include <hip/hip_runtime.h>
#include <stdint.h>

#define NIMG     128
#define HW       65536          // 256*256
#define THREADS  1024
#define GPT      16             // float4 groups per thread (16*4 = 64 pixels)
#define KM_ITERS 20

typedef __attribute__((ext_vector_type(2))) float v2f;
typedef __attribute__((ext_vector_type(8))) float v8f;

// ---- block-wide reductions (wave32 shfl + 32-slot LDS scratch) ----
__device__ __forceinline__ float blockSum(float v, volatile float* scratch) {
    __syncthreads();                       // protect scratch reuse across calls
#pragma unroll
    for (int m = 16; m >= 1; m >>= 1) v += __shfl_xor(v, m, 32);
    if ((threadIdx.x & 31) == 0) scratch[threadIdx.x >> 5] = v;
    __syncthreads();
    float r = scratch[threadIdx.x & 31];   // 32 waves -> 32 slots
#pragma unroll
    for (int m = 16; m >= 1; m >>= 1) r += __shfl_xor(r, m, 32);
    return r;                              // total in every thread
}

__device__ __forceinline__ float blockMin(float v, volatile float* scratch) {
    __syncthreads();
#pragma unroll
    for (int m = 16; m >= 1; m >>= 1) v = fminf(v, __shfl_xor(v, m, 32));
    if ((threadIdx.x & 31) == 0) scratch[threadIdx.x >> 5] = v;
    __syncthreads();
    float r = scratch[threadIdx.x & 31];
#pragma unroll
    for (int m = 16; m >= 1; m >>= 1) r = fminf(r, __shfl_xor(r, m, 32));
    return r;
}

__device__ __forceinline__ float blockMax(float v, volatile float* scratch) {
    __syncthreads();
#pragma unroll
    for (int m = 16; m >= 1; m >>= 1) v = fmaxf(v, __shfl_xor(v, m, 32));
    if ((threadIdx.x & 31) == 0) scratch[threadIdx.x >> 5] = v;
    __syncthreads();
    float r = scratch[threadIdx.x & 31];
#pragma unroll
    for (int m = 16; m >= 1; m >>= 1) r = fmaxf(r, __shfl_xor(r, m, 32));
    return r;
}

// One workgroup per image. Gray image lives entirely in VGPRs (64 f32/thread).
// sr_mask image is async-staged into LDS (CDNA5 global->LDS async path) while
// the 20 register-resident k-means iterations run.
__global__ __launch_bounds__(THREADS)
void kmeans_mask_loss_kernel(const float* __restrict__ hr,
                             const float* __restrict__ sr,
                             float* __restrict__ part) {
    extern __shared__ float lds_sr[];      // HW floats (256 KB dynamic LDS)
    __shared__ float red[32];

    const int b = blockIdx.x;
    const int t = threadIdx.x;
    const float* img = hr + (size_t)b * 3u * HW;
    const float* srb = sr + (size_t)b * HW;

    const unsigned lds_base = (unsigned)(uintptr_t)(void*)lds_sr;

    // ---- 1) kick off async staging of sr_mask -> LDS (16 B per lane per op) ----
#pragma unroll
    for (int i = 0; i < GPT; ++i) {
        const unsigned g = (unsigned)t + (unsigned)i * THREADS;   // float4 group id
        const unsigned loff = lds_base + g * 16u;
        const unsigned long long gaddr =
            (unsigned long long)(uintptr_t)(srb + 4u * g);
        asm volatile("global_load_async_to_lds_b128 %0, %1, off"
                     :: "v"(loff), "v"(gaddr) : "memory");
    }

    // ---- 2) load 3 channels once, gray -> registers; local min/max/sum ----
    float gray[64];
    float vmin = 3.402823466e38f, vmax = -3.402823466e38f, lsum = 0.f;
#pragma unroll
    for (int i = 0; i < GPT; ++i) {
        const unsigned g = (unsigned)t + (unsigned)i * THREADS;
        const float4 r4 = *(const float4*)(img + 4u * g);
        const float4 g4 = *(const float4*)(img + HW + 4u * g);
        const float4 b4 = *(const float4*)(img + 2u * HW + 4u * g);
        float gx = (r4.x + g4.x + b4.x) * (1.f / 3.f);
        float gy = (r4.y + g4.y + b4.y) * (1.f / 3.f);
        float gz = (r4.z + g4.z + b4.z) * (1.f / 3.f);
        float gw = (r4.w + g4.w + b4.w) * (1.f / 3.f);
        gray[i * 4 + 0] = gx; gray[i * 4 + 1] = gy;
        gray[i * 4 + 2] = gz; gray[i * 4 + 3] = gw;
        vmin = fminf(vmin, fminf(fminf(gx, gy), fminf(gz, gw)));
        vmax = fmaxf(vmax, fmaxf(fmaxf(gx, gy), fmaxf(gz, gw)));
        lsum += (gx + gy) + (gz + gw);
    }
    float c0 = blockMin(vmin, red);        // deterministic init: c0=min, c1=max
    float c1 = blockMax(vmax, red);
    const float stot = blockSum(lsum, red);
    const float ntot = (float)HW;

    // ---- 3) 20 Lloyd iterations, all data in registers ----
    for (int it = 0; it < KM_ITERS; ++it) {
        float s1 = 0.f, n1 = 0.f;
#pragma unroll
        for (int e = 0; e < 64; ++e) {
            const float x = gray[e];
            // jnp.argmin tie-break: cluster 1 only on strict <
            if (fabsf(x - c1) < fabsf(x - c0)) { s1 += x; n1 += 1.f; }
        }
        s1 = blockSum(s1, red);
        n1 = blockSum(n1, red);
        const float n0 = ntot - n1;
        const float s0 = stot - s1;
        c0 = s0 / fmaxf(n0, 1.f);
        c1 = s1 / fmaxf(n1, 1.f);
    }

    // ---- 4) final assignment + border votes ----
    uint32_t cb0 = 0, cb1 = 0;
    float fc = 0.f, lc = 0.f, fr = 0.f, lr = 0.f;
#pragma unroll
    for (int i = 0; i < GPT; ++i) {
        const unsigned g = (unsigned)t + (unsigned)i * THREADS;
#pragma unroll
        for (int j = 0; j < 4; ++j) {
            const int e = i * 4 + j;
            const float x = gray[e];
            const bool one = fabsf(x - c1) < fabsf(x - c0);
            if (one) { if (e < 32) cb0 |= 1u << e; else cb1 |= 1u << (e - 32); }
            const unsigned p   = 4u * g + (unsigned)j;
            const unsigned col = p & 255u;
            const unsigned row = p >> 8;
            const float o = one ? 1.f : 0.f;
            if (col == 0u)   fc += o;
            if (col == 255u) lc += o;
            if (row == 0u)   fr += o;
            if (row == 255u) lr += o;
        }
    }
    fc = blockSum(fc, red);
    lc = blockSum(lc, red);
    fr = blockSum(fr, red);
    lr = blockSum(lr, red);
    const int num = (fr > 128.f) + (lr > 128.f) + (fc > 128.f) + (lc > 128.f);
    const bool flip = (num >= 3);

    // ---- 5) wait for async sr staging, then MSE partial ----
    asm volatile("s_wait_asynccnt 0" ::: "memory");
    __syncthreads();

    float acc = 0.f;
#pragma unroll
    for (int i = 0; i < GPT; ++i) {
        const unsigned g = (unsigned)t + (unsigned)i * THREADS;
        const float4 s4 = *(const float4*)(lds_sr + 4u * g);
#pragma unroll
        for (int j = 0; j < 4; ++j) {
            const int e = i * 4 + j;
            const uint32_t bit = (e < 32) ? ((cb0 >> e) & 1u) : ((cb1 >> (e - 32)) & 1u);
            float m = bit ? 1.f : 0.f;
            if (flip) m = 1.f - m;
            const float sv = (j == 0) ? s4.x : (j == 1) ? s4.y : (j == 2) ? s4.z : s4.w;
            const float d = sv - m;
            acc += d * d;
        }
    }
    acc = blockSum(acc, red);
    if (t == 0) part[b] = acc;
}

// Reduce 128 per-image partials with two chained v_wmma_f32_16x16x4_f32
// (B = ones => D[m,n] = sum_k A[m,k] + C ; exact f32 K-reduction), one wave.
__global__ __launch_bounds__(32)
void final_reduce_wmma(const float* __restrict__ part, float* __restrict__ out) {
    const int L  = threadIdx.x;            // 0..31
    const int m  = L & 15;
    const int kb = (L < 16) ? 0 : 2;       // 16x4 f32 A layout: v0=K0/K2, v1=K1/K3
    v2f a0, a1, bones;
    a0.x = part[4 * m + kb + 0];
    a0.y = part[4 * m + kb + 1];
    a1.x = part[64 + 4 * m + kb + 0];
    a1.y = part[64 + 4 * m + kb + 1];
    bones.x = 1.f; bones.y = 1.f;
    v8f c = {0.f, 0.f, 0.f, 0.f, 0.f, 0.f, 0.f, 0.f};
    c = __builtin_amdgcn_wmma_f32_16x16x4_f32(false, a0, false, bones, (short)0, c, false, false);
    c = __builtin_amdgcn_wmma_f32_16x16x4_f32(false, a1, false, bones, (short)0, c, false, false);
    // lane L<16 holds D rows 0..7 (all N-columns equal); lanes>=16 rows 8..15
    float tsum = ((c[0] + c[1]) + (c[2] + c[3])) + ((c[4] + c[5]) + (c[6] + c[7]));
    tsum += __shfl_xor(tsum, 16, 32);      // rows 0..7 + rows 8..15 = grand total
    if (L == 0) out[0] = tsum * (1.0f / 8388608.0f);   // mean over 128*256*256
}

extern "C" void kernel_launch(void* const* d_in, const int* in_sizes, int n_in,
                              void* d_out, int out_size, void* d_ws, size_t ws_size,
                              hipStream_t stream) {
    const float* hr = (const float*)d_in[0];   // [128,3,256,256] f32
    const float* sr = (const float*)d_in[1];   // [128,1,256,256] f32
    float* out  = (float*)d_out;               // scalar f32
    float* part = (float*)d_ws;                // 128 per-image partial sums

    kmeans_mask_loss_kernel<<<NIMG, THREADS, HW * sizeof(float), stream>>>(hr, sr, part);
    final_reduce_wmma<<<1, 32, 0, stream>>>(part, out);
}